// MixtureRouting_69741678952492
// MI455X (gfx1250) — compile-verified
//
#include <hip/hip_runtime.h>
#include <hip/hip_bf16.h>

// Problem constants (match reference)
#define BB 4
#define DD 81
#define KK 2048
// derived tiling for GEMM: M tiles of 16 over D=81 -> 6 tiles (rows 81..95 unused)
#define MTILES 6
#define NGROUPS (KK / 64)   // 32 groups of 64 columns (4 x 16-wide WMMA tiles)

typedef __attribute__((ext_vector_type(2))) float v2f;
typedef __attribute__((ext_vector_type(8))) float v8f;

// ---------------------------------------------------------------------------
// Kernel A: build coef[b, p, k] = lh / sum_k lh  (pi^4 factor cancels exactly)
// One 256-thread workgroup per (b, p) row; 8 elements per thread.
// ---------------------------------------------------------------------------
__global__ void __launch_bounds__(256)
mixroute_coef_kernel(const float* __restrict__ pi,
                     const float* __restrict__ mu,
                     const float* __restrict__ sig,
                     float* __restrict__ coef)
{
    const int b = blockIdx.x >> 11;        // / KK
    const int p = blockIdx.x & (KK - 1);

    const float* __restrict__ muB  = mu  + (size_t)b * 4 * KK;
    const float* __restrict__ sigB = sig + (size_t)b * 4 * KK;
    const float* __restrict__ piB  = pi  + (size_t)b * KK;

    // point coordinates = component-p means (row constants)
    const float mp0 = muB[p];
    const float mp1 = muB[KK + p];
    const float mp2 = muB[2 * KK + p];
    const float mp3 = muB[3 * KK + p];

    const int tid = threadIdx.x;

    float lh[8];
    float s = 0.f;
#pragma unroll
    for (int i = 0; i < 8; ++i) {
        const int k = tid + i * 256;
        const float s0 = sigB[k];
        const float s1 = sigB[KK + k];
        const float s2 = sigB[2 * KK + k];
        const float s3 = sigB[3 * KK + k];
        const float d0 = mp0 - muB[k];
        const float d1 = mp1 - muB[KK + k];
        const float d2 = mp2 - muB[2 * KK + k];
        const float d3 = mp3 - muB[3 * KK + k];
        const float num = piB[k] * s0 * s1 * s2 * s3;
        const float den = (d0 * d0 + s0 * s0) * (d1 * d1 + s1 * s1) *
                          (d2 * d2 + s2 * s2) * (d3 * d3 + s3 * s3);
        const float v = num / den;          // 1/pi^4 omitted: cancels in normalization
        lh[i] = v;
        s += v;
    }

    // wave32 butterfly reduction, then cross-wave via LDS
#pragma unroll
    for (int off = 16; off > 0; off >>= 1)
        s += __shfl_xor(s, off, 32);

    __shared__ float red[8];
    if ((tid & 31) == 0) red[tid >> 5] = s;
    __syncthreads();

    float tot = 0.f;
#pragma unroll
    for (int w = 0; w < 8; ++w) tot += red[w];
    const float inv = 1.f / tot;

    float* __restrict__ out = coef + (size_t)b * KK * KK + (size_t)p * KK;
#pragma unroll
    for (int i = 0; i < 8; ++i)
        out[tid + i * 256] = lh[i] * inv;
}

// ---------------------------------------------------------------------------
// Kernel B: y[b] = x[b] (81 x 2048) * coef[b] (2048 x 2048) via fp32 WMMA.
// One wave computes a 16(M) x 64(N) output tile: 4 accumulators reuse the A
// fragment. 768 wave-tasks -> 96 blocks x 8 waves.
//
// V_WMMA_F32_16X16X4_F32 fragment layouts (wave32):
//   A (16x4):  lanes 0-15: M=lane,   v0=K0, v1=K1 ; lanes 16-31: M=lane-16, v0=K2, v1=K3
//   B (4x16):  lanes 0-15: N=lane,   v0=K0, v1=K1 ; lanes 16-31: N=lane-16, v0=K2, v1=K3
//   C/D (16x16): VGPR r: lanes 0-15 -> M=r, N=lane ; lanes 16-31 -> M=8+r, N=lane-16
//
// OOB handling: WMMA output row M == A row M, so A rows >= 81 only influence
// output rows that are never stored. Clamp the A row for the load (no fault,
// no EXEC churn in the hot loop) and predicate only the final stores.
// ---------------------------------------------------------------------------
__global__ void __launch_bounds__(256)
mixroute_gemm_kernel(const float* __restrict__ X,
                     const float* __restrict__ C,
                     float* __restrict__ Y)
{
    const int wave = (int)((blockIdx.x * blockDim.x + threadIdx.x) >> 5);
    const int b     = wave / (MTILES * NGROUPS);
    const int rem   = wave % (MTILES * NGROUPS);
    const int mtile = rem / NGROUPS;
    const int ngrp  = rem % NGROUPS;
    const int n0    = ngrp * 64;

    const int lane = threadIdx.x & 31;
    const int half = lane >> 4;      // 0: K pair (0,1), 1: K pair (2,3)
    const int l15  = lane & 15;

    const int arow       = mtile * 16 + l15;
    const int arow_clamp = (arow < DD) ? arow : (DD - 1);   // safe address, value unused if OOB
    const float* __restrict__ Xp = X + (size_t)b * DD * KK + (size_t)arow_clamp * KK + half * 2;
    const float* __restrict__ Cb = C + (size_t)b * KK * KK + (size_t)half * 2 * KK + n0 + l15;

    v8f acc0 = {}, acc1 = {}, acc2 = {}, acc3 = {};

#pragma unroll 4
    for (int kk = 0; kk < KK; kk += 4) {
        v2f a;
        a.x = Xp[kk];
        a.y = Xp[kk + 1];

        const float* __restrict__ Crow = Cb + (size_t)kk * KK;  // row kk(+half*2); next row at +KK

        v2f b0, b1, b2, b3;
        b0.x = Crow[0];     b0.y = Crow[KK];
        b1.x = Crow[16];    b1.y = Crow[KK + 16];
        b2.x = Crow[32];    b2.y = Crow[KK + 32];
        b3.x = Crow[48];    b3.y = Crow[KK + 48];

        acc0 = __builtin_amdgcn_wmma_f32_16x16x4_f32(false, a, false, b0, (short)0, acc0, false, false);
        acc1 = __builtin_amdgcn_wmma_f32_16x16x4_f32(false, a, false, b1, (short)0, acc1, false, false);
        acc2 = __builtin_amdgcn_wmma_f32_16x16x4_f32(false, a, false, b2, (short)0, acc2, false, false);
        acc3 = __builtin_amdgcn_wmma_f32_16x16x4_f32(false, a, false, b3, (short)0, acc3, false, false);
    }

    float* __restrict__ Yb = Y + (size_t)b * DD * KK;
#pragma unroll
    for (int r = 0; r < 8; ++r) {
        const int row = mtile * 16 + half * 8 + r;
        if (row < DD) {
            float* __restrict__ yr = Yb + (size_t)row * KK + n0 + l15;
            yr[0]  = acc0[r];
            yr[16] = acc1[r];
            yr[32] = acc2[r];
            yr[48] = acc3[r];
        }
    }
}

// ---------------------------------------------------------------------------
extern "C" void kernel_launch(void* const* d_in, const int* in_sizes, int n_in,
                              void* d_out, int out_size, void* d_ws, size_t ws_size,
                              hipStream_t stream)
{
    const float* x   = (const float*)d_in[0];   // (B, D, K)
    const float* pi  = (const float*)d_in[1];   // (B, 1, K)
    const float* mu  = (const float*)d_in[2];   // (B, 4, K)
    const float* sig = (const float*)d_in[3];   // (B, 4, K)

    float* coef = (float*)d_ws;                 // (B, K, K) = 67 MB scratch
    float* y    = (float*)d_out;                // (B, D, K)

    // Stage 1: coefficient matrix (one block per (b, p) row)
    mixroute_coef_kernel<<<BB * KK, 256, 0, stream>>>(pi, mu, sig, coef);

    // Stage 2: batched GEMM via fp32 WMMA (768 wave tasks / 8 waves per block)
    const int total_waves = BB * MTILES * NGROUPS;   // 768
    mixroute_gemm_kernel<<<total_waves / 8, 256, 0, stream>>>(x, coef, y);
}